// CapsMaxPool_2894807957973
// MI455X (gfx1250) — compile-verified
//
#include <hip/hip_runtime.h>
#include <stdint.h>

// Capsule 2x2 max-pool by L2 norm, x:[64,64,64,32,8] f32 -> out:[64,32,32,32,8] f32.
// Pure streaming op (AI ~0.4 flop/B): bound by HBM at ~14.4us on MI455X (320 MiB @ 23.3 TB/s).
// gfx1250 path: async global->LDS staging (ASYNCcnt) + b128 coalesced accesses + NT stores.

typedef float v4f __attribute__((ext_vector_type(4)));
typedef int   v4i __attribute__((vector_size(16)));   // matches builtin param pointee

namespace {
constexpr int Bn = 64, Hn = 64, Wn = 64, Cn = 32, An = 8;
constexpr int HoN = Hn / 2, WoN = Wn / 2;
constexpr int PIX = Bn * HoN * WoN;                    // 65536 output pixels
constexpr long long ROW = (long long)Wn * Cn * An;     // 16384 floats per input row
constexpr int PXF = Cn * An;                           // 256 floats per (b,h,w) pixel
}

__device__ __forceinline__ float sqsum8(v4f a, v4f b) {
  float s = a[0] * a[0];
  s = fmaf(a[1], a[1], s);
  s = fmaf(a[2], a[2], s);
  s = fmaf(a[3], a[3], s);
  s = fmaf(b[0], b[0], s);
  s = fmaf(b[1], b[1], s);
  s = fmaf(b[2], b[2], s);
  s = fmaf(b[3], b[3], s);
  return s;
}

__global__ __launch_bounds__(256) void caps_maxpool_2894(const float* __restrict__ x,
                                                         float* __restrict__ out) {
  const int lane = threadIdx.x & 31;   // lane == channel c (wave32 == C)
  const int wib  = threadIdx.x >> 5;   // wave within block (8 waves/block)
  const int p    = (int)blockIdx.x * 8 + wib;  // output pixel id = (b*Ho+ho)*Wo+wo
  const int wo = p & (WoN - 1);
  const int ho = (p >> 5) & (HoN - 1);
  const int b  = p >> 10;
  // float offset of window origin (b, 2*ho, 2*wo)
  const long long base = ((long long)(b * Hn + 2 * ho) * Wn + 2 * wo) * PXF;

  v4f h0[4], h1[4];  // 4 candidate capsules (2 x float4 each), s = ph*2+pw

#if __has_builtin(__builtin_amdgcn_global_load_async_to_lds_b128) && \
    __has_builtin(__builtin_amdgcn_s_wait_asynccnt)
  // ---- gfx1250 async-tensor path: stage 4 x 1KiB window segments into this
  // wave's private LDS slice; tracked by ASYNCcnt, no VGPRs held by loads.
  typedef __attribute__((address_space(1))) v4i* gp_t;  // global (device) ptr
  typedef __attribute__((address_space(3))) v4i* lp_t;  // LDS ptr
  __shared__ v4f tile[8][4 * 64];  // 8 waves * 4 KiB = 32 KiB/block
  v4f* lsl = tile[wib];
  #pragma unroll
  for (int s = 0; s < 4; ++s) {
    const float* g = x + base + (long long)(s >> 1) * ROW + (s & 1) * PXF;
    #pragma unroll
    for (int j = 0; j < 2; ++j) {  // 2 x (32 lanes * 16B) = 1024B per segment
      const float* gp = g + (j * 32 + lane) * 4;
      v4f* lp = &lsl[s * 64 + j * 32 + lane];
      __builtin_amdgcn_global_load_async_to_lds_b128(
          (gp_t)(uintptr_t)gp,
          (lp_t)(uint32_t)(uintptr_t)lp,
          0, 0);
    }
  }
  __builtin_amdgcn_s_wait_asynccnt(0);  // this wave's LDS writes landed
  __asm__ volatile("" ::: "memory");
  #pragma unroll
  for (int s = 0; s < 4; ++s) {
    h0[s] = lsl[s * 64 + 2 * lane + 0];
    h1[s] = lsl[s * 64 + 2 * lane + 1];
  }
#else
  // ---- fallback: direct coalesced b128 NT loads (still roofline-optimal)
  #pragma unroll
  for (int s = 0; s < 4; ++s) {
    const v4f* g =
        (const v4f*)(x + base + (long long)(s >> 1) * ROW + (s & 1) * PXF) + 2 * lane;
    h0[s] = __builtin_nontemporal_load(g);
    h1[s] = __builtin_nontemporal_load(g + 1);
  }
#endif

  // Squared L2 norms (argmax-equivalent to norms; first-index tie-break via strict >)
  const float n0 = sqsum8(h0[0], h1[0]);
  const float n1 = sqsum8(h0[1], h1[1]);
  const float n2 = sqsum8(h0[2], h1[2]);
  const float n3 = sqsum8(h0[3], h1[3]);

  float bn = n0;
  v4f w0 = h0[0], w1 = h1[0];
  {
    const bool t = n1 > bn;
    bn = t ? n1 : bn; w0 = t ? h0[1] : w0; w1 = t ? h1[1] : w1;
  }
  {
    const bool t = n2 > bn;
    bn = t ? n2 : bn; w0 = t ? h0[2] : w0; w1 = t ? h1[2] : w1;
  }
  {
    const bool t = n3 > bn;
    bn = t ? n3 : bn; w0 = t ? h0[3] : w0; w1 = t ? h1[3] : w1;
  }
  (void)bn;

  // out pixel p, channel = lane: 32B per lane -> 1KiB contiguous per wave; NT store
  v4f* o = (v4f*)(out + (long long)p * PXF) + 2 * lane;
  __builtin_nontemporal_store(w0, o);
  __builtin_nontemporal_store(w1, o + 1);
}

extern "C" void kernel_launch(void* const* d_in, const int* in_sizes, int n_in,
                              void* d_out, int out_size, void* d_ws, size_t ws_size,
                              hipStream_t stream) {
  (void)in_sizes; (void)n_in; (void)out_size; (void)d_ws; (void)ws_size;
  const float* x = (const float*)d_in[0];
  float* out = (float*)d_out;
  // one wave per output pixel: 65536 pixels / 8 waves per block = 8192 blocks
  caps_maxpool_2894<<<dim3(PIX / 8), dim3(256), 0, stream>>>(x, out);
}